// DenseLSTM_23502061044059
// MI455X (gfx1250) — compile-verified
//
#include <hip/hip_runtime.h>

typedef float v2f __attribute__((ext_vector_type(2)));
typedef float v8f __attribute__((ext_vector_type(8)));

#define SEQ_LEN 131072
#define HID 64
#define G4 256  // 4*HID

// ---------------------------------------------------------------------------
// Kernel 1: x_proj[t][g] = inputs[t][:] . W_ih[g][:] + b_ih[g] + b_hh[g]
// One wave per 16-row block: A tile cached in registers once, then looped
// over the 16 column tiles (16x less A traffic than tile-per-wave).
// K=64 via 16 chained f32 16x16x4 WMMAs per column tile.
// ---------------------------------------------------------------------------
__global__ __launch_bounds__(256) void xproj_wmma(
    const float* __restrict__ inputs, const float* __restrict__ W_ih,
    const float* __restrict__ b_ih, const float* __restrict__ b_hh,
    float* __restrict__ x_proj)
{
  const int lane = threadIdx.x & 31;
  const int wave = threadIdx.x >> 5;
  const int rt   = blockIdx.x * 8 + wave;   // 8192 row tiles (T/16)
  const int l15  = lane & 15;
  const int half = lane >> 4;
  const int row0 = rt * 16;

  // Cache the 16x64 A tile: f32 A layout -> lane l15 = M, half selects K pair.
  const v2f* Arow = reinterpret_cast<const v2f*>(inputs + (size_t)(row0 + l15) * HID);
  v2f areg[16];
#pragma unroll
  for (int kk = 0; kk < 16; ++kk) areg[kk] = Arow[kk * 2 + half];

  for (int ct = 0; ct < 16; ++ct) {
    const int  col  = ct * 16 + l15;
    const v2f* Brow = reinterpret_cast<const v2f*>(W_ih + (size_t)col * HID);
    v8f c = {};
#pragma unroll
    for (int kk = 0; kk < 16; ++kk) {
      v2f b = Brow[kk * 2 + half];   // B[k][n] = W_ih[n][k]
      c = __builtin_amdgcn_wmma_f32_16x16x4_f32(false, areg[kk], false, b,
                                                (short)0, c, false, false);
    }
    const float bias = b_ih[col] + b_hh[col];
#pragma unroll
    for (int v = 0; v < 8; ++v) {
      const int row = row0 + v + half * 8;   // C layout: lanes16-31 -> M = v+8
      x_proj[(size_t)row * G4 + col] = c[v] + bias;
    }
  }
}

// ---------------------------------------------------------------------------
// Kernel 2: sequential LSTM scan. One 256-thread workgroup owns the chain.
// Thread j keeps W_hh row j in 64 VGPRs; h is broadcast from LDS as float4.
// Dot product uses 4 independent accumulators (dependent chain = 16 FMAs,
// not 64), and x_proj is prefetched 2 steps ahead to cover L2 latency.
// ---------------------------------------------------------------------------
__device__ __forceinline__ float fsig(float x)  { return 1.0f / (1.0f + __expf(-x)); }
__device__ __forceinline__ float ftanh(float x) { return 1.0f - 2.0f / (__expf(2.0f * x) + 1.0f); }

__global__ __launch_bounds__(256) void lstm_scan(
    const float* __restrict__ x_proj, const float* __restrict__ W_hh,
    float* __restrict__ hs)
{
  __shared__ float sh_h[HID];
  __shared__ float sh_z[G4];
  const int j = threadIdx.x;

  float w[HID];
#pragma unroll
  for (int k = 0; k < HID; ++k) w[k] = W_hh[(size_t)j * HID + k];

  float creg = 0.0f;
  if (j < HID) sh_h[j] = 0.0f;
  __syncthreads();

  // Depth-2 software pipeline on the x_proj stream.
  float xp0 = x_proj[j];                       // t = 0
  float xp1 = x_proj[(size_t)G4 + j];          // t = 1

  for (int t = 0; t < SEQ_LEN; ++t) {
    const int   tn  = (t + 2 < SEQ_LEN) ? (t + 2) : (SEQ_LEN - 1);
    const float xp2 = x_proj[(size_t)tn * G4 + j];   // prefetch t+2

    // z_j = xp + W_hh[j,:] . h   with 4 independent FMA chains
    float a0 = xp0, a1 = 0.0f, a2 = 0.0f, a3 = 0.0f;
    const float4* h4 = reinterpret_cast<const float4*>(sh_h);
#pragma unroll
    for (int k = 0; k < 4; ++k) {
      const float4 h0 = h4[4 * k + 0];
      const float4 h1 = h4[4 * k + 1];
      const float4 h2 = h4[4 * k + 2];
      const float4 h3 = h4[4 * k + 3];
      const int b = 16 * k;
      a0 = fmaf(w[b +  0], h0.x, a0); a0 = fmaf(w[b +  1], h0.y, a0);
      a0 = fmaf(w[b +  2], h0.z, a0); a0 = fmaf(w[b +  3], h0.w, a0);
      a1 = fmaf(w[b +  4], h1.x, a1); a1 = fmaf(w[b +  5], h1.y, a1);
      a1 = fmaf(w[b +  6], h1.z, a1); a1 = fmaf(w[b +  7], h1.w, a1);
      a2 = fmaf(w[b +  8], h2.x, a2); a2 = fmaf(w[b +  9], h2.y, a2);
      a2 = fmaf(w[b + 10], h2.z, a2); a2 = fmaf(w[b + 11], h2.w, a2);
      a3 = fmaf(w[b + 12], h3.x, a3); a3 = fmaf(w[b + 13], h3.y, a3);
      a3 = fmaf(w[b + 14], h3.z, a3); a3 = fmaf(w[b + 15], h3.w, a3);
    }
    sh_z[j] = (a0 + a1) + (a2 + a3);
    __syncthreads();

    if (j < HID) {
      const float ig = fsig (sh_z[0 * HID + j]);
      const float fg = fsig (sh_z[1 * HID + j]);
      const float gg = ftanh(sh_z[2 * HID + j]);
      const float og = fsig (sh_z[3 * HID + j]);
      creg = fg * creg + ig * gg;
      const float hn = og * ftanh(creg);
      sh_h[j] = hn;
      hs[(size_t)t * HID + j] = hn;  // fire-and-forget store
    }
    __syncthreads();
    xp0 = xp1;
    xp1 = xp2;
  }
}

// ---------------------------------------------------------------------------
// Kernel 3: fused head. One wave per 16-row block:
//   u = relu(h); v = relu(u @ W_lin^T + b_lin); y = v . W_fin + b_fin
// A cached in registers, 4 col-tiles x 16 WMMAs, shfl_xor column reduction.
// ---------------------------------------------------------------------------
__global__ __launch_bounds__(256) void head_wmma(
    const float* __restrict__ hs, const float* __restrict__ W_lin,
    const float* __restrict__ b_lin, const float* __restrict__ W_fin,
    const float* __restrict__ b_fin, float* __restrict__ out)
{
  const int lane = threadIdx.x & 31;
  const int wave = threadIdx.x >> 5;
  const int rt   = blockIdx.x * 8 + wave;  // 8192 row tiles
  const int l15  = lane & 15;
  const int half = lane >> 4;
  const int row0 = rt * 16;

  // Load + relu the 16x64 A tile once into registers.
  const v2f* Arow = reinterpret_cast<const v2f*>(hs + (size_t)(row0 + l15) * HID);
  v2f areg[16];
#pragma unroll
  for (int kk = 0; kk < 16; ++kk) {
    v2f a = Arow[kk * 2 + half];
    a.x = fmaxf(a.x, 0.0f);
    a.y = fmaxf(a.y, 0.0f);
    areg[kk] = a;
  }

  float partial[8];
#pragma unroll
  for (int v = 0; v < 8; ++v) partial[v] = 0.0f;

#pragma unroll
  for (int ctile = 0; ctile < 4; ++ctile) {
    const int  col  = ctile * 16 + l15;
    const v2f* Brow = reinterpret_cast<const v2f*>(W_lin + (size_t)col * HID);
    v8f c = {};
#pragma unroll
    for (int kk = 0; kk < 16; ++kk) {
      v2f b = Brow[kk * 2 + half];
      c = __builtin_amdgcn_wmma_f32_16x16x4_f32(false, areg[kk], false, b,
                                                (short)0, c, false, false);
    }
    const float bl = b_lin[col];
    const float wf = W_fin[col];
#pragma unroll
    for (int v = 0; v < 8; ++v)
      partial[v] += fmaxf(c[v] + bl, 0.0f) * wf;
  }

  // Reduce over the 16 column-lanes of each half (C layout: N = l15).
#pragma unroll
  for (int v = 0; v < 8; ++v) {
    float s = partial[v];
    s += __shfl_xor(s, 1, 32);
    s += __shfl_xor(s, 2, 32);
    s += __shfl_xor(s, 4, 32);
    s += __shfl_xor(s, 8, 32);
    if (l15 == 0) out[row0 + half * 8 + v] = s + b_fin[0];
  }
}

// ---------------------------------------------------------------------------
extern "C" void kernel_launch(void* const* d_in, const int* in_sizes, int n_in,
                              void* d_out, int out_size, void* d_ws, size_t ws_size,
                              hipStream_t stream) {
  const float* inputs = (const float*)d_in[0];
  const float* W_ih   = (const float*)d_in[1];
  const float* W_hh   = (const float*)d_in[2];
  const float* b_ih   = (const float*)d_in[3];
  const float* b_hh   = (const float*)d_in[4];
  const float* W_lin  = (const float*)d_in[5];
  const float* b_lin  = (const float*)d_in[6];
  const float* W_fin  = (const float*)d_in[7];
  const float* b_fin  = (const float*)d_in[8];
  float* out = (float*)d_out;

  float* x_proj = (float*)d_ws;                       // SEQ_LEN*256 f32 = 134 MB
  float* hs     = x_proj + (size_t)SEQ_LEN * G4;      // SEQ_LEN*64  f32 =  34 MB

  // T/16 = 8192 row tiles, 8 waves/block -> 1024 blocks
  xproj_wmma<<<1024, 256, 0, stream>>>(inputs, W_ih, b_ih, b_hh, x_proj);
  // single workgroup owns the sequential recurrence
  lstm_scan<<<1, 256, 0, stream>>>(x_proj, W_hh, hs);
  // T/16 = 8192 row tiles, 8 waves/block -> 1024 blocks
  head_wmma<<<1024, 256, 0, stream>>>(hs, W_lin, b_lin, W_fin, b_fin, out);
}